// CTCLoss_59519656788385
// MI455X (gfx1250) — compile-verified
//
#include <hip/hip_runtime.h>

// CTC loss, MI455X (gfx1250).
// Shapes fixed by the reference: T=512, B=32, V=8000, L=100, S=2L+1=201.
#define T_DIM 512
#define B_DIM 32
#define V_DIM 8000
#define L_DIM 100
#define S_DIM 201
#define SP    208            // padded S stride
#define NEGF  (-1e30f)

#define AS1 __attribute__((address_space(1)))
#define AS3 __attribute__((address_space(3)))

typedef int v4i __attribute__((__vector_size__(4 * sizeof(int))));

// ---- CDNA5 async global->LDS copy (ASYNCcnt-tracked) -----------------------
__device__ __forceinline__ void async_copy_b128(const void* gsrc, void* lds_dst) {
#if __has_builtin(__builtin_amdgcn_global_load_async_to_lds_b128)
  __builtin_amdgcn_global_load_async_to_lds_b128(
      (AS1 v4i*)gsrc, (AS3 v4i*)lds_dst, /*offset=*/0, /*cpol=*/0);
#else
  unsigned lds_addr = (unsigned)(unsigned long long)(AS3 void*)lds_dst;
  unsigned long long gaddr = (unsigned long long)gsrc;
  asm volatile("global_load_async_to_lds_b128 %0, %1, off"
               :: "v"(lds_addr), "v"(gaddr) : "memory");
#endif
}

__device__ __forceinline__ void wait_asynccnt0() {
#if __has_builtin(__builtin_amdgcn_s_wait_asynccnt)
  __builtin_amdgcn_s_wait_asynccnt(0);
#else
  asm volatile("s_wait_asynccnt 0" ::: "memory");
#endif
}

// ---- Kernel 1: per-(t,b) log-softmax row -> gathered extended log-probs ----
// One block per row (t*B+b). Stage the 32KB row in LDS via async DMA (single
// HBM read), reduce max then sum(exp), then write lp_ext[s] = row[ext[s]]-lse.
__global__ __launch_bounds__(256) void ctc_row_lse_gather(
    const float* __restrict__ acts, const int* __restrict__ targets,
    float* __restrict__ lp_ext) {
  __shared__ float buf[V_DIM];
  __shared__ float red[8];
  __shared__ float sval;

  const int tid  = threadIdx.x;
  const int row  = blockIdx.x;           // row = t*B + b
  const int b    = row % B_DIM;
  const float* grow = acts + (size_t)row * V_DIM;

  // 8000 floats = 2000 x 16B chunks, coalesced async copies.
#pragma unroll
  for (int j = 0; j < 8; ++j) {
    int c = tid + 256 * j;
    if (c < V_DIM / 4) async_copy_b128(grow + c * 4, buf + c * 4);
  }
  wait_asynccnt0();
  __syncthreads();

  const int lane = tid & 31;
  const int wid  = tid >> 5;

  // ---- max reduce ----
  float m = NEGF;
  for (int i = tid; i < V_DIM; i += 256) m = fmaxf(m, buf[i]);
#pragma unroll
  for (int o = 16; o > 0; o >>= 1) m = fmaxf(m, __shfl_xor(m, o, 32));
  if (lane == 0) red[wid] = m;
  __syncthreads();
  if (tid == 0) {
    float mm = red[0];
#pragma unroll
    for (int i = 1; i < 8; ++i) mm = fmaxf(mm, red[i]);
    sval = mm;
  }
  __syncthreads();
  m = sval;

  // ---- sum(exp(x-m)) reduce ----
  float s = 0.f;
  for (int i = tid; i < V_DIM; i += 256) s += __expf(buf[i] - m);
#pragma unroll
  for (int o = 16; o > 0; o >>= 1) s += __shfl_xor(s, o, 32);
  if (lane == 0) red[wid] = s;
  __syncthreads();
  if (tid == 0) {
    float ss = red[0];
#pragma unroll
    for (int i = 1; i < 8; ++i) ss += red[i];
    sval = m + __logf(ss);               // lse
  }
  __syncthreads();
  const float lse = sval;

  // ---- fused gather: lp_ext[s] = logp[ext[s]] straight from LDS ----
  if (tid < S_DIM) {
    int e = (tid & 1) ? targets[b * L_DIM + (tid >> 1)] : 0;  // blank at even s
    lp_ext[(size_t)row * SP + tid] = buf[e] - lse;
  }
}

// ---- Kernel 2: forward alpha recursion, one wave32 per batch sample --------
// alpha in LDS double buffer with 2-slot NEG padding; 7 states per lane.
// Single-wave workgroup: barriers lower to S_NOP (cheap), LDS is in-order.
__global__ __launch_bounds__(32) void ctc_scan(
    const float* __restrict__ lp_ext, const int* __restrict__ targets,
    const int* __restrict__ act_lens, const int* __restrict__ label_lens,
    float* __restrict__ loss) {
  __shared__ float alp[2][S_DIM + 2];

  const int lane = threadIdx.x;
  const int b    = blockIdx.x;
  const float* lpb = lp_ext + (size_t)b * SP;
  const size_t stride = (size_t)B_DIM * SP;   // between consecutive t rows

  // can_skip[s] = (ext[s]!=blank) && (ext[s]!=ext[s-2]); even s -> blank -> false
  bool cs[7];
#pragma unroll
  for (int k = 0; k < 7; ++k) {
    int s = lane + 32 * k;
    bool c = false;
    if (s < S_DIM && (s & 1)) {
      int idx = s >> 1;
      int e   = targets[b * L_DIM + idx];
      int em2 = (idx > 0) ? targets[b * L_DIM + idx - 1] : -1;
      c = (e != 0) && (e != em2);
    }
    cs[k] = c;
  }

  for (int i = lane; i < S_DIM + 2; i += 32) { alp[0][i] = NEGF; alp[1][i] = NEGF; }
  __syncthreads();
  if (lane == 0) { alp[0][2] = lpb[0]; alp[0][3] = lpb[1]; }
  __syncthreads();

  const int alen = act_lens[b];
  int pp = 0;

  // prefetch lp row for t=1
  float cur[7];
#pragma unroll
  for (int k = 0; k < 7; ++k) {
    int s = lane + 32 * k;
    cur[k] = (s < S_DIM) ? lpb[stride + s] : 0.f;
  }

  for (int t = 1; t < T_DIM; ++t) {
    // prefetch row t+1 while computing t
    float nxt[7];
#pragma unroll
    for (int k = 0; k < 7; ++k) {
      int s = lane + 32 * k;
      nxt[k] = (t + 1 < T_DIM && s < S_DIM) ? lpb[(size_t)(t + 1) * stride + s] : 0.f;
    }

    const bool active = (t < alen);
    float res[7];
#pragma unroll
    for (int k = 0; k < 7; ++k) {
      int s = lane + 32 * k;
      if (s < S_DIM) {
        float a1 = alp[pp][s + 2];
        float a2 = alp[pp][s + 1];
        float a3 = cs[k] ? alp[pp][s] : NEGF;
        float mm = fmaxf(a1, fmaxf(a2, a3));
        float r  = mm + __logf(__expf(a1 - mm) + __expf(a2 - mm) + __expf(a3 - mm));
        res[k] = active ? (r + cur[k]) : a1;   // freeze past act_len
      }
    }
#pragma unroll
    for (int k = 0; k < 7; ++k) {
      int s = lane + 32 * k;
      if (s < S_DIM) alp[1 - pp][s + 2] = res[k];
    }
    pp ^= 1;
    __syncthreads();   // single-wave WG: S_NOP-class barrier, orders LDS
#pragma unroll
    for (int k = 0; k < 7; ++k) cur[k] = nxt[k];
  }

  if (lane == 0) {
    int   sl = 2 * label_lens[b];
    float x  = alp[pp][sl + 2];
    float y  = alp[pp][sl + 1];
    float mm = fmaxf(x, y);
    loss[b]  = -(mm + __logf(__expf(x - mm) + __expf(y - mm)));
  }
}

// ---- Kernel 3: mean over batch --------------------------------------------
__global__ __launch_bounds__(32) void ctc_finalize(const float* __restrict__ loss,
                                                   float* __restrict__ out) {
  float v = (threadIdx.x < B_DIM) ? loss[threadIdx.x] : 0.f;
#pragma unroll
  for (int o = 16; o > 0; o >>= 1) v += __shfl_xor(v, o, 32);
  if (threadIdx.x == 0) out[0] = v / (float)B_DIM;
}

extern "C" void kernel_launch(void* const* d_in, const int* in_sizes, int n_in,
                              void* d_out, int out_size, void* d_ws, size_t ws_size,
                              hipStream_t stream) {
  const float* acts       = (const float*)d_in[0];   // [T,B,V] f32
  const int*   targets    = (const int*)d_in[1];     // [B,L] i32
  const int*   act_lens   = (const int*)d_in[2];     // [B] i32
  const int*   label_lens = (const int*)d_in[3];     // [B] i32

  float* lp_ext = (float*)d_ws;                            // T*B*SP floats (~13.6 MB)
  float* loss   = lp_ext + (size_t)T_DIM * B_DIM * SP;     // B floats

  ctc_row_lse_gather<<<T_DIM * B_DIM, 256, 0, stream>>>(acts, targets, lp_ext);
  ctc_scan<<<B_DIM, 32, 0, stream>>>(lp_ext, targets, act_lens, label_lens, loss);
  ctc_finalize<<<1, 32, 0, stream>>>(loss, (float*)d_out);
}